// feature_selection_node_53858889892405
// MI455X (gfx1250) — compile-verified
//
#include <hip/hip_runtime.h>

// ---------------------------------------------------------------------------
// feature_selection: sigmoid -> top-200 of 1000 per tree -> scatter into
// (64,16080) attention -> out[b,t,f] = x[b,f] * att[t,f]  (256,64,16080)
//
// Store-bound problem (~1.06 GB written, 23.3 TB/s HBM => ~45us floor).
// Strategy: B128 non-temporal streaming stores; 93.8% of output is a pure
// zero-fill; inputs staged via CDNA5 async global->LDS (ASYNCcnt) path, with
// the async wait deferred past the zero-fill stream to hide load latency.
// ---------------------------------------------------------------------------

typedef float v4f __attribute__((ext_vector_type(4)));

#define T_TREES  64
#define N_ATT    1000
#define K_TOP    200
#define FEAT     16080
#define B_BATCH  256
#define RV_ELEMS ((size_t)B_BATCH * (size_t)T_TREES * (size_t)FEAT)  // 263,454,720

// ---- CDNA5 async global->LDS helpers (gfx1250, tracked by ASYNCcnt) -------

__device__ __forceinline__ unsigned lds_byte_addr(const void* p) {
  // generic -> LDS(as3) pointer -> 32-bit LDS byte offset
  return (unsigned)(size_t)(__attribute__((address_space(3))) const void*)p;
}

__device__ __forceinline__ void async_g2l_b128(unsigned lds_byte,
                                               unsigned gvoff_bytes,
                                               const void* sbase) {
  // GVS mode: 64-bit SGPR base + 32-bit per-lane VGPR byte offset.
  // VDST vgpr carries the LDS byte address.
  asm volatile("global_load_async_to_lds_b128 %0, %1, %2"
               :
               : "v"(lds_byte), "v"(gvoff_bytes), "s"(sbase)
               : "memory");
}

__device__ __forceinline__ void wait_async0() {
  asm volatile("s_wait_asynccnt 0" ::: "memory");
}

// ---------------------------------------------------------------------------
// Kernel 1: per-tree sigmoid + exact top-K select (radix select on fp32 bits,
// valid since sigmoid outputs are strictly positive) + scatter row write.
// One block per tree; 256 threads.
// ---------------------------------------------------------------------------
__global__ void __launch_bounds__(256)
build_attention_kernel(const float* __restrict__ mask,
                       float* __restrict__ att_out) {
  __shared__ float    vals[N_ATT];
  __shared__ unsigned hist[256];
  __shared__ unsigned sh_sel[2];  // [0]=bit prefix of kth value, [1]=remaining K among ties

  const int t   = blockIdx.x;
  const int tid = threadIdx.x;

  for (int i = tid; i < N_ATT; i += 256) {
    float m = mask[t * N_ATT + i];
    vals[i] = 1.0f / (1.0f + expf(-m));
  }
  if (tid == 0) { sh_sel[0] = 0u; sh_sel[1] = K_TOP; }
  __syncthreads();

  // 4-pass MSB-first radix select: find exact bit pattern of the K-th largest.
  for (int pass = 0; pass < 4; ++pass) {
    const int shift = 24 - 8 * pass;
    hist[tid] = 0u;
    __syncthreads();
    const unsigned pre    = sh_sel[0];
    const unsigned himask = (pass == 0) ? 0u : (0xFFFFFFFFu << (shift + 8));
    for (int i = tid; i < N_ATT; i += 256) {
      unsigned b = __float_as_uint(vals[i]);
      if ((b & himask) == (pre & himask))
        atomicAdd(&hist[(b >> shift) & 0xFFu], 1u);
    }
    __syncthreads();
    if (tid == 0) {
      unsigned remk = sh_sel[1];
      unsigned cum  = 0;
      int d = 255;
      for (; d > 0; --d) {              // walk bins from largest value down
        if (cum + hist[d] >= remk) break;
        cum += hist[d];
      }
      sh_sel[0] = pre | ((unsigned)d << shift);
      sh_sel[1] = remk - cum;
    }
    __syncthreads();
  }

  const unsigned kth  = sh_sel[0];
  const unsigned remk = sh_sel[1];

  float* orow = att_out + (size_t)t * FEAT;
  for (int f = tid; f < FEAT; f += 256) {
    float v = 0.0f;
    if (f < N_ATT) {
      float s = vals[f];
      unsigned b = __float_as_uint(s);
      if (b > kth) {
        v = s;
      } else if (b == kth) {
        // tie-break by lowest index, matching lax.top_k
        unsigned r = 0;
        for (int j = 0; j < f; ++j)
          if (__float_as_uint(vals[j]) == kth) ++r;
        if (r < remk) v = s;
      }
    }
    orow[f] = v;  // regular (RT) stores: kernel 2 re-reads these rows from L2
  }
}

// ---------------------------------------------------------------------------
// Kernel 2: one block per (b,t) output row of 16080 floats.
//   1) issue async B128 global->LDS copies of att[t,0:1000] and x[b,0:1000]
//   2) stream the structural-zero region f in [1000,16080) with NT B128
//      stores (independent work that hides the async-load latency)
//   3) s_wait_asynccnt 0, multiply, NT B128 store f in [0,1000)
// ---------------------------------------------------------------------------
__global__ void __launch_bounds__(256)
broadcast_mul_kernel(const float* __restrict__ x,
                     const float* __restrict__ att,
                     float* __restrict__ out) {
  __shared__ v4f lbuf[512];  // [0..249] = att chunk, [256..505] = x chunk

  const int row = blockIdx.x;
  const int b   = row >> 6;
  const int t   = row & 63;
  const int tid = threadIdx.x;

  const float* arow = att + (size_t)t * FEAT;
  const float* xrow = x   + (size_t)b * FEAT;
  float*       orow = out + (size_t)row * FEAT;

  // (1) kick off async global->LDS copies (ASYNCcnt path)
  if (tid < (N_ATT / 4)) {  // 250 lanes, 16B each = 4000B per row chunk
    const unsigned goff = (unsigned)tid * 16u;
    async_g2l_b128(lds_byte_addr(&lbuf[tid]),       goff, (const void*)arow);
    async_g2l_b128(lds_byte_addr(&lbuf[256 + tid]), goff, (const void*)xrow);
  }

  // (2) zero-fill f = 1000 .. 16079 : 3770 float4 NT stores per row,
  //     fully independent of the async loads -> hides their latency
  v4f z = 0.0f;
  for (int i = tid; i < (FEAT - N_ATT) / 4; i += 256) {
    __builtin_nontemporal_store(z, (v4f*)(orow + N_ATT + i * 4));
  }

  // (3) consume the staged tiles; each lane reads only the 16B it requested,
  //     so s_wait_asynccnt 0 suffices (no workgroup barrier needed)
  wait_async0();
  if (tid < (N_ATT / 4)) {
    v4f a  = lbuf[tid];
    v4f xv = lbuf[256 + tid];
    v4f r  = a * xv;
    __builtin_nontemporal_store(r, (v4f*)(orow + tid * 4));
  }
}

// ---------------------------------------------------------------------------

extern "C" void kernel_launch(void* const* d_in, const int* in_sizes, int n_in,
                              void* d_out, int out_size, void* d_ws, size_t ws_size,
                              hipStream_t stream) {
  const float* x    = (const float*)d_in[0];  // (256,120,134) contiguous == (256,16080)
  const float* mask = (const float*)d_in[1];  // (64,1000)
  float* out = (float*)d_out;                 // [RV (256,64,16080) | attention (64,16080)]
  float* att = out + RV_ELEMS;

  build_attention_kernel<<<dim3(T_TREES), dim3(256), 0, stream>>>(mask, att);
  broadcast_mul_kernel<<<dim3(B_BATCH * T_TREES), dim3(256), 0, stream>>>(x, att, out);
}